// GNet_63419487093236
// MI455X (gfx1250) — compile-verified
//
#include <hip/hip_runtime.h>
#include <math.h>

// ---------------------------------------------------------------------------
// GCN value head for MI455X (gfx1250, wave32).
// GEMMs use V_WMMA_F32_16X16X4_F32 (exact fp32 matrix pipe).
// Edge aggregation uses coalesced global_atomic_add_f32 (L2-resident, 192MB L2).
// ---------------------------------------------------------------------------

typedef float v2f __attribute__((ext_vector_type(2)));
typedef float v8f __attribute__((ext_vector_type(8)));

#define H 32
#define P 1024

// ---------------- simple elementwise kernels ----------------

__global__ void k_fill(float* __restrict__ p, float v, int n) {
  int i = blockIdx.x * blockDim.x + threadIdx.x;
  if (i < n) p[i] = v;
}

__global__ void k_deg_edges(const int* __restrict__ dst, float* __restrict__ deg, int E) {
  int i = blockIdx.x * blockDim.x + threadIdx.x;
  if (i < E) {
    __hip_atomic_fetch_add(&deg[dst[i]], 1.0f, __ATOMIC_RELAXED, __HIP_MEMORY_SCOPE_AGENT);
  }
}

__global__ void k_rsqrt(float* __restrict__ p, int n) {
  int i = blockIdx.x * blockDim.x + threadIdx.x;
  if (i < n) p[i] = rsqrtf(p[i]);
}

// out[n][c] = h[n][c] * dis[n]^2   (self-loop term, also initializes accumulator)
__global__ void k_self(const float* __restrict__ h, const float* __restrict__ dis,
                       float* __restrict__ out, int total /* N*32 */) {
  int i = blockIdx.x * blockDim.x + threadIdx.x;
  if (i < total) {
    int n = i >> 5;
    float d = dis[n];
    out[i] = h[i] * d * d;
  }
}

// out[dst][c] += h[src][c] * dis[src]*dis[dst]
// One wave == one edge: lanes 0..31 are the 32 features -> src/dst loads are
// wave-uniform, gather + atomic are single coalesced 128B transactions.
__global__ void k_scatter(const float* __restrict__ h, const float* __restrict__ dis,
                          const int* __restrict__ src, const int* __restrict__ dst,
                          float* __restrict__ out, int E) {
  int i = blockIdx.x * blockDim.x + threadIdx.x;
  int e = i >> 5;
  int c = i & 31;
  if (e < E) {
    int s = src[e];
    int d = dst[e];
    float nrm = dis[s] * dis[d];
    float val = h[s * H + c] * nrm;
    __hip_atomic_fetch_add(&out[d * H + c], val, __ATOMIC_RELAXED, __HIP_MEMORY_SCOPE_AGENT);
  }
}

// out[n][c] = tanh(in[n][c] + bias[c])
__global__ void k_bias_tanh(const float* __restrict__ in, const float* __restrict__ bias,
                            float* __restrict__ out, int total) {
  int i = blockIdx.x * blockDim.x + threadIdx.x;
  if (i < total) {
    out[i] = tanhf(in[i] + bias[i & 31]);
  }
}

// ---------------- WMMA fp32 GEMM: Y[N,32] = X[N,K] @ W[32,K]^T ----------------
// One wave computes one 16x16 output tile via V_WMMA_F32_16X16X4_F32, K in steps of 4.
// A layout: lane l (0..15 / 16..31), .x/.y = K = {kk + 2*(lane/16), +1}, row M = m0 + (lane&15)
// B layout: lane covers col n0+(lane&15), same K split.
// C/D: element v -> row m0 + v + 8*(lane/16), col n0 + (lane&15).
__global__ void k_wmma_gemm_nt(const float* __restrict__ X, const float* __restrict__ W,
                               float* __restrict__ Y, int K) {
  int gtid = blockIdx.x * blockDim.x + threadIdx.x;
  int wave = gtid >> 5;
  int lane = threadIdx.x & 31;
  int half = lane >> 4;     // 0 or 1
  int l    = lane & 15;

  int m0 = (wave >> 1) * 16;       // row tile
  int n0 = (wave & 1) * 16;        // col tile (32 cols -> 2 tiles)

  const float* xrow = X + (size_t)(m0 + l) * K;
  const float* wrow = W + (size_t)(n0 + l) * K;

  v8f acc = {};
  for (int kk = 0; kk < K; kk += 4) {
    v2f a = *(const v2f*)(xrow + kk + 2 * half);
    v2f b = *(const v2f*)(wrow + kk + 2 * half);
    acc = __builtin_amdgcn_wmma_f32_16x16x4_f32(
        /*neg_a=*/false, a, /*neg_b=*/false, b,
        /*c_mod=*/(short)0, acc, /*reuse_a=*/false, /*reuse_b=*/false);
  }

  float* yout = Y + (size_t)(m0 + 8 * half) * H + n0 + l;
#pragma unroll
  for (int v = 0; v < 8; ++v) yout[(size_t)v * H] = acc[v];
}

// ---------------- global add pool: pooled[b][c] = sum_{r<1024} h[b*1024+r][c] ----
__global__ void k_pool(const float* __restrict__ h, float* __restrict__ pooled) {
  __shared__ float lds[256];
  int b = blockIdx.x;
  int t = threadIdx.x;          // 256 threads
  int c = t & 31;
  int r0 = t >> 5;              // 0..7
  float s = 0.0f;
  const float* base = h + (size_t)b * P * H;
  for (int r = r0; r < P; r += 8) s += base[r * H + c];
  lds[t] = s;
  __syncthreads();
  if (t < H) {
    float acc = 0.0f;
#pragma unroll
    for (int j = 0; j < 8; ++j) acc += lds[c + 32 * j];
    pooled[b * H + c] = acc;
  }
}

// ---------------- fused head: Wp, value_net MLP, value_comnet -> out[B] -------
__global__ void k_head(const float* __restrict__ pooled, const float* __restrict__ share,
                       const float* __restrict__ Wp, const float* __restrict__ bp,
                       const float* __restrict__ Vw1, const float* __restrict__ Vb1,
                       const float* __restrict__ Vw2, const float* __restrict__ Vb2,
                       const float* __restrict__ Vw3, const float* __restrict__ Vb3,
                       const float* __restrict__ Cw1, const float* __restrict__ Cb1,
                       const float* __restrict__ Cw2, const float* __restrict__ Cb2,
                       float* __restrict__ out, int B) {
  int b = threadIdx.x;
  if (b >= B) return;

  float h1[H], h2[H], t[H];

  // h1 = pooled[b] @ Wp^T + bp   (no activation, per reference)
#pragma unroll 4
  for (int o = 0; o < H; ++o) {
    float s = bp[o];
    for (int k = 0; k < H; ++k) s += pooled[b * H + k] * Wp[o * H + k];
    h1[o] = s;
  }
  // h2 = tanh(share @ Vw1^T + Vb1)
#pragma unroll 4
  for (int o = 0; o < H; ++o) {
    float s = Vb1[o];
    for (int k = 0; k < 64; ++k) s += share[b * 64 + k] * Vw1[o * 64 + k];
    t[o] = tanhf(s);
  }
  // tanh(t @ Vw2^T + Vb2)
#pragma unroll 4
  for (int o = 0; o < H; ++o) {
    float s = Vb2[o];
    for (int k = 0; k < H; ++k) s += t[k] * Vw2[o * H + k];
    h2[o] = tanhf(s);
  }
  // h2final = h2 @ Vw3^T + Vb3   (no activation)
#pragma unroll 4
  for (int o = 0; o < H; ++o) {
    float s = Vb3[o];
    for (int k = 0; k < H; ++k) s += h2[k] * Vw3[o * H + k];
    t[o] = s;
  }
  // value = tanh(concat(h1, t) @ Cw1^T + Cb1) @ Cw2^T + Cb2
  float val = Cb2[0];
#pragma unroll 4
  for (int o = 0; o < H; ++o) {
    float s = Cb1[o];
    for (int k = 0; k < H; ++k) s += h1[k] * Cw1[o * 64 + k];
    for (int k = 0; k < H; ++k) s += t[k] * Cw1[o * 64 + 32 + k];
    val += tanhf(s) * Cw2[o];
  }
  out[b] = val;
}

// ---------------------------------------------------------------------------

extern "C" void kernel_launch(void* const* d_in, const int* in_sizes, int n_in,
                              void* d_out, int out_size, void* d_ws, size_t ws_size,
                              hipStream_t stream) {
  const float* x     = (const float*)d_in[0];
  const int*   eidx  = (const int*)d_in[1];
  const float* share = (const float*)d_in[3];
  const float* W1 = (const float*)d_in[4];
  const float* b1 = (const float*)d_in[5];
  const float* W2 = (const float*)d_in[6];
  const float* b2 = (const float*)d_in[7];
  const float* Wl = (const float*)d_in[8];
  const float* bl = (const float*)d_in[9];
  const float* Wp = (const float*)d_in[10];
  const float* bp = (const float*)d_in[11];
  const float* Vw1 = (const float*)d_in[12];
  const float* Vb1 = (const float*)d_in[13];
  const float* Vw2 = (const float*)d_in[14];
  const float* Vb2 = (const float*)d_in[15];
  const float* Vw3 = (const float*)d_in[16];
  const float* Vb3 = (const float*)d_in[17];
  const float* Cw1 = (const float*)d_in[18];
  const float* Cb1 = (const float*)d_in[19];
  const float* Cw2 = (const float*)d_in[20];
  const float* Cb2 = (const float*)d_in[21];
  float* out = (float*)d_out;

  const int N = in_sizes[0] / 16;     // 65536
  const int E = in_sizes[1] / 2;      // 2M
  const int B = in_sizes[3] / 64;     // 64
  const int D_IN = 16;
  const int NH = N * H;

  const int* src = eidx;
  const int* dst = eidx + E;

  // workspace layout (floats)
  float* dis    = (float*)d_ws;                 // N
  float* bufA   = dis + N;                      // N*H
  float* bufB   = bufA + (size_t)N * H;         // N*H
  float* pooled = bufB + (size_t)N * H;         // B*H

  const int T = 256;
  dim3 blk(T);
  auto g = [&](long long n) { return dim3((unsigned)((n + T - 1) / T)); };

  // degree -> dis = rsqrt(deg)   (self-loop gives deg >= 1)
  k_fill<<<g(N), blk, 0, stream>>>(dis, 1.0f, N);
  k_deg_edges<<<g(E), blk, 0, stream>>>(dst, dis, E);
  k_rsqrt<<<g(N), blk, 0, stream>>>(dis, N);

  const int gemm_blocks = (N / 16) * 2 * 32 / T;   // one wave per 16x16 tile

  // ---- conv1: hA = tanh(Agg(X @ W1^T) + b1)
  k_wmma_gemm_nt<<<gemm_blocks, blk, 0, stream>>>(x, W1, bufA, D_IN);
  k_self<<<g(NH), blk, 0, stream>>>(bufA, dis, bufB, NH);
  k_scatter<<<g((long long)E * H), blk, 0, stream>>>(bufA, dis, src, dst, bufB, E);
  k_bias_tanh<<<g(NH), blk, 0, stream>>>(bufB, b1, bufA, NH);

  // ---- conv2: hB = tanh(Agg(hA @ W2^T) + b2)
  k_wmma_gemm_nt<<<gemm_blocks, blk, 0, stream>>>(bufA, W2, bufB, H);
  k_self<<<g(NH), blk, 0, stream>>>(bufB, dis, bufA, NH);
  k_scatter<<<g((long long)E * H), blk, 0, stream>>>(bufB, dis, src, dst, bufA, E);
  k_bias_tanh<<<g(NH), blk, 0, stream>>>(bufA, b2, bufB, NH);

  // ---- linear: hA = tanh(hB @ Wl^T + bl)
  k_wmma_gemm_nt<<<gemm_blocks, blk, 0, stream>>>(bufB, Wl, bufA, H);
  k_bias_tanh<<<g(NH), blk, 0, stream>>>(bufA, bl, bufA, NH);

  // ---- pool + head
  k_pool<<<dim3(B), blk, 0, stream>>>(bufA, pooled);
  k_head<<<dim3(1), dim3(64), 0, stream>>>(pooled, share,
                                           Wp, bp, Vw1, Vb1, Vw2, Vb2, Vw3, Vb3,
                                           Cw1, Cb1, Cw2, Cb2, out, B);
}